// LSTM_7859790152288
// MI455X (gfx1250) — compile-verified
//
#include <hip/hip_runtime.h>
#include <cstdint>

// ---------------------------------------------------------------------------
// 2-layer LSTM (B=256, T=1200, I=16, H=128) + linear head, fp32 throughout.
//
// MI455X / gfx1250 (wave32) strategy:
//  * One persistent workgroup per 16-row batch tile (16 WGs x 512 threads =
//    16 waves). Each WG scans all T=1200 steps. Recurrent weights W_hh^T are
//    pinned in LDS for the whole kernel in a k-pair-interleaved layout so
//    every WMMA B-fragment is a single aligned b64 LDS load. Input-GEMM
//    weights (W_ih) live in VGPRs as preloaded WMMA B-fragments.
//  * All matmuls use V_WMMA_F32_16X16X4_F32 (fp32) to preserve precision
//    across the 1200-step recurrence. Accumulators start at 0 (inline C=0);
//    biases are folded into the gate-update phase from registers.
//  * Per-step input staging (x tile / h1 tile) uses the CDNA5 async path:
//    global_load_async_to_lds_b32/b128 + s_wait_asynccnt, issued after the
//    mid-step barrier so it overlaps the DS/VALU tail of the step.
//  * Layer0 writes h1[B,T,H] to workspace; layer1 consumes it and fuses
//    relu + W_lin dot (wave shuffle reduction) into the output.
// ---------------------------------------------------------------------------

typedef float v2f __attribute__((ext_vector_type(2)));
typedef float v8f __attribute__((ext_vector_type(8)));

#define B_TOT 256
#define T_LEN 1200
#define I_DIM 16
#define H_DIM 128
#define G_DIM 512   // 4*H
#define BT    16    // batch tile per workgroup

// LDS strides (floats), chosen for conflict-free fragment access
#define WP   1056   // interleaved W^T row stride: 64 rows x [2n+(k&1)]; 1056%64==32
#define HSTR 132    // h rows    (132%64==4)
#define GSTR 524    // gate rows (524%64==12, 8*524%64==32)
#define XSTR 20     // x rows

__device__ __forceinline__ float fast_sigmoid(float x) {
  return 1.0f / (1.0f + __expf(-x));
}
__device__ __forceinline__ float fast_tanh(float x) {
  float ax = fabsf(x);
  float e  = __expf(-2.0f * ax);           // e in (0,1], no overflow
  float r  = (1.0f - e) / (1.0f + e);
  return copysignf(r, x);
}

// D = A(16x4) * B(4x16) + C, fp32
__device__ __forceinline__ v8f wmma4(v2f a, v2f b, v8f c) {
  return __builtin_amdgcn_wmma_f32_16x16x4_f32(false, a, false, b, (short)0, c,
                                               false, false);
}

// CDNA5 async global->LDS copies (tracked by ASYNCcnt)
__device__ __forceinline__ void async_copy_b32(uint32_t lds_off, const void* g) {
  asm volatile("global_load_async_to_lds_b32 %0, %1, off"
               :: "v"(lds_off), "v"(g) : "memory");
}
__device__ __forceinline__ void async_copy_b128(uint32_t lds_off, const void* g) {
  asm volatile("global_load_async_to_lds_b128 %0, %1, off"
               :: "v"(lds_off), "v"(g) : "memory");
}
__device__ __forceinline__ void wait_async0() {
  asm volatile("s_wait_asynccnt 0" ::: "memory");
}

__device__ __forceinline__ uint32_t lds_off(const void* p) {
  return (uint32_t)(uintptr_t)p;   // low 32 bits of flat addr = LDS offset
}

// ===========================================================================
// Layer 0: x[B,T,16] -> h1[B,T,128]
// ===========================================================================
__global__ __launch_bounds__(512, 1) void lstm_layer0(
    const float* __restrict__ x, const float* __restrict__ Wih,
    const float* __restrict__ Whh, const float* __restrict__ bih,
    const float* __restrict__ bhh, float* __restrict__ h1out) {
  __shared__ float sW[64 * WP];        // W_hh^T interleaved  270336 B
  __shared__ float sX[BT * XSTR];      // x_t tile              1280 B
  __shared__ float sH[BT * HSTR];      // h state               8448 B
  __shared__ float sG[BT * GSTR];      // gate tile            33536 B
  __shared__ float sBias[G_DIM];       //                       2048 B

  const int tid  = threadIdx.x;
  const int lane = tid & 31;
  const int wave = tid >> 5;           // 0..15
  const int hsel = lane >> 4;          // lane half (0/1) -> K split
  const int l16  = lane & 15;          // column within tile
  const int b0   = blockIdx.x * BT;
  const int n0   = wave * 32;          // this wave's two 16-col tiles

  // --- one-time fills: W^T into k-pair-interleaved layout -------------------
  for (int i = tid; i < G_DIM * H_DIM; i += 512) {
    const int n = i >> 7, k = i & (H_DIM - 1);
    sW[(k >> 1) * WP + 2 * n + (k & 1)] = Whh[i];
  }
  if (tid < G_DIM) sBias[tid] = bih[tid] + bhh[tid];
  for (int i = tid; i < BT * HSTR; i += 512) sH[i] = 0.0f;

  // Preload W_ih0 B-fragments (K=16 -> 4 k-frags, 2 tiles) into VGPRs.
  v2f bi[2][4];
#pragma unroll
  for (int tt = 0; tt < 2; ++tt) {
#pragma unroll
    for (int kf = 0; kf < 4; ++kf) {
      const int n = n0 + tt * 16 + l16;
      const int k = kf * 4 + hsel * 2;
      const float2 w = *reinterpret_cast<const float2*>(&Wih[n * I_DIM + k]);
      bi[tt][kf].x = w.x;
      bi[tt][kf].y = w.y;
    }
  }

  float cc0 = 0.f, cc1 = 0.f, cc2 = 0.f, cc3 = 0.f;  // cell state (regs)
  const int ub = tid >> 5;            // update phase: batch row 0..15
  const int uj = (tid & 31) * 4;      // update phase: 4 hidden cols

  // stage x for t = 0 (async to LDS)
  if (tid < 256) {
    const int bb = tid >> 4, k = tid & 15;
    async_copy_b32(lds_off(&sX[bb * XSTR + k]),
                   &x[((size_t)(b0 + bb) * T_LEN + 0) * I_DIM + k]);
  }
  wait_async0();
  __syncthreads();

  // time-invariant gate biases for this thread's 4 hidden columns
  const float4 bI = *reinterpret_cast<const float4*>(&sBias[0 * H_DIM + uj]);
  const float4 bF = *reinterpret_cast<const float4*>(&sBias[1 * H_DIM + uj]);
  const float4 bG = *reinterpret_cast<const float4*>(&sBias[2 * H_DIM + uj]);
  const float4 bO = *reinterpret_cast<const float4*>(&sBias[3 * H_DIM + uj]);

  for (int t = 0; t < T_LEN; ++t) {
    // ---- g = x_t @ Wih^T + h @ Whh^T  (two 16x16 tiles per wave) ----------
    v8f acc0 = {};
    v8f acc1 = {};

#pragma unroll
    for (int kf = 0; kf < 4; ++kf) {            // input part, K=16
      const int k = kf * 4 + hsel * 2;
      const v2f a = *reinterpret_cast<const v2f*>(&sX[l16 * XSTR + k]);
      acc0 = wmma4(a, bi[0][kf], acc0);
      acc1 = wmma4(a, bi[1][kf], acc1);
    }
#pragma unroll
    for (int kf = 0; kf < 32; ++kf) {           // recurrent part, K=128
      const int k = kf * 4 + hsel * 2;
      const v2f a  = *reinterpret_cast<const v2f*>(&sH[l16 * HSTR + k]);
      const int wr = (kf * 2 + hsel) * WP;
      const v2f w0 = *reinterpret_cast<const v2f*>(&sW[wr + 2 * (n0 + l16)]);
      const v2f w1 = *reinterpret_cast<const v2f*>(&sW[wr + 2 * (n0 + 16 + l16)]);
      acc0 = wmma4(a, w0, acc0);
      acc1 = wmma4(a, w1, acc1);
    }
    // spill gate tiles to LDS (row m = r + 8*lanehalf, col = n0+l16)
#pragma unroll
    for (int r = 0; r < 8; ++r) {
      const int m = r + hsel * 8;
      sG[m * GSTR + n0 + l16]      = acc0[r];
      sG[m * GSTR + n0 + 16 + l16] = acc1[r];
    }
    __syncthreads();

    // ---- gate nonlinearity + state update (4 elements per thread) ---------
    float4 gi = *reinterpret_cast<const float4*>(&sG[ub * GSTR + 0 * H_DIM + uj]);
    float4 gf = *reinterpret_cast<const float4*>(&sG[ub * GSTR + 1 * H_DIM + uj]);
    float4 gg = *reinterpret_cast<const float4*>(&sG[ub * GSTR + 2 * H_DIM + uj]);
    float4 go = *reinterpret_cast<const float4*>(&sG[ub * GSTR + 3 * H_DIM + uj]);
    gi.x += bI.x; gi.y += bI.y; gi.z += bI.z; gi.w += bI.w;
    gf.x += bF.x; gf.y += bF.y; gf.z += bF.z; gf.w += bF.w;
    gg.x += bG.x; gg.y += bG.y; gg.z += bG.z; gg.w += bG.w;
    go.x += bO.x; go.y += bO.y; go.z += bO.z; go.w += bO.w;
    float4 hv;
    {
      float i_ = fast_sigmoid(gi.x), f_ = fast_sigmoid(gf.x);
      float g_ = fast_tanh(gg.x),    o_ = fast_sigmoid(go.x);
      cc0 = f_ * cc0 + i_ * g_;  hv.x = o_ * fast_tanh(cc0);
    }
    {
      float i_ = fast_sigmoid(gi.y), f_ = fast_sigmoid(gf.y);
      float g_ = fast_tanh(gg.y),    o_ = fast_sigmoid(go.y);
      cc1 = f_ * cc1 + i_ * g_;  hv.y = o_ * fast_tanh(cc1);
    }
    {
      float i_ = fast_sigmoid(gi.z), f_ = fast_sigmoid(gf.z);
      float g_ = fast_tanh(gg.z),    o_ = fast_sigmoid(go.z);
      cc2 = f_ * cc2 + i_ * g_;  hv.z = o_ * fast_tanh(cc2);
    }
    {
      float i_ = fast_sigmoid(gi.w), f_ = fast_sigmoid(gf.w);
      float g_ = fast_tanh(gg.w),    o_ = fast_sigmoid(go.w);
      cc3 = f_ * cc3 + i_ * g_;  hv.w = o_ * fast_tanh(cc3);
    }
    *reinterpret_cast<float4*>(&sH[ub * HSTR + uj]) = hv;
    *reinterpret_cast<float4*>(&h1out[((size_t)(b0 + ub) * T_LEN + t) * H_DIM + uj]) = hv;

    // stage x for t+1 (sX reads of step t all completed before mid-barrier)
    if (t + 1 < T_LEN && tid < 256) {
      const int bb = tid >> 4, k = tid & 15;
      async_copy_b32(lds_off(&sX[bb * XSTR + k]),
                     &x[((size_t)(b0 + bb) * T_LEN + (t + 1)) * I_DIM + k]);
    }
    wait_async0();
    __syncthreads();
  }
}

// ===========================================================================
// Layer 1 + head: h1[B,T,128] -> out[B,T,1]
// ===========================================================================
__global__ __launch_bounds__(512, 1) void lstm_layer1_out(
    const float* __restrict__ h1, const float* __restrict__ Wih,
    const float* __restrict__ Whh, const float* __restrict__ bih,
    const float* __restrict__ bhh, const float* __restrict__ Wlin,
    const float* __restrict__ blin, float* __restrict__ out) {
  __shared__ float sW[64 * WP];        // W_hh1^T interleaved 270336 B
  __shared__ float sHin[BT * HSTR];    // layer-0 h_t tile      8448 B
  __shared__ float sH[BT * HSTR];      // layer-1 h state       8448 B
  __shared__ float sG[BT * GSTR];      //                      33536 B
  __shared__ float sBias[G_DIM];       //                       2048 B
  __shared__ float sWlin[H_DIM];       //                        512 B

  const int tid  = threadIdx.x;
  const int lane = tid & 31;
  const int wave = tid >> 5;
  const int hsel = lane >> 4;
  const int l16  = lane & 15;
  const int b0   = blockIdx.x * BT;
  const int n0   = wave * 32;

  for (int i = tid; i < G_DIM * H_DIM; i += 512) {
    const int n = i >> 7, k = i & (H_DIM - 1);
    sW[(k >> 1) * WP + 2 * n + (k & 1)] = Whh[i];
  }
  if (tid < G_DIM) sBias[tid] = bih[tid] + bhh[tid];
  if (tid < H_DIM) sWlin[tid] = Wlin[tid];
  for (int i = tid; i < BT * HSTR; i += 512) sH[i] = 0.0f;

  // Preload W_ih1 B-fragments into VGPRs: 2 tiles x 32 k-frags = 128 VGPRs.
  v2f bi[2][32];
#pragma unroll
  for (int tt = 0; tt < 2; ++tt) {
#pragma unroll
    for (int kf = 0; kf < 32; ++kf) {
      const int n = n0 + tt * 16 + l16;
      const int k = kf * 4 + hsel * 2;
      const float2 w = *reinterpret_cast<const float2*>(&Wih[n * H_DIM + k]);
      bi[tt][kf].x = w.x;
      bi[tt][kf].y = w.y;
    }
  }

  float cc0 = 0.f, cc1 = 0.f, cc2 = 0.f, cc3 = 0.f;
  const int ub = tid >> 5;
  const int uj = (tid & 31) * 4;
  const float blv = blin[0];

  // stage h1_t for t = 0 (async b128 to LDS)
  {
    const int bb = tid >> 5, j = (tid & 31) * 4;
    async_copy_b128(lds_off(&sHin[bb * HSTR + j]),
                    &h1[((size_t)(b0 + bb) * T_LEN + 0) * H_DIM + j]);
  }
  wait_async0();
  __syncthreads();

  const float4 bI = *reinterpret_cast<const float4*>(&sBias[0 * H_DIM + uj]);
  const float4 bF = *reinterpret_cast<const float4*>(&sBias[1 * H_DIM + uj]);
  const float4 bG = *reinterpret_cast<const float4*>(&sBias[2 * H_DIM + uj]);
  const float4 bO = *reinterpret_cast<const float4*>(&sBias[3 * H_DIM + uj]);

  for (int t = 0; t < T_LEN; ++t) {
    v8f acc0 = {};
    v8f acc1 = {};

    // input part: h1_t @ Wih1^T, K=128, B-frags from registers (full unroll)
#pragma unroll
    for (int kf = 0; kf < 32; ++kf) {
      const int k = kf * 4 + hsel * 2;
      const v2f a = *reinterpret_cast<const v2f*>(&sHin[l16 * HSTR + k]);
      acc0 = wmma4(a, bi[0][kf], acc0);
      acc1 = wmma4(a, bi[1][kf], acc1);
    }
    // recurrent part: h2 @ Whh1^T, K=128, B-frags from LDS (b64 each)
#pragma unroll 8
    for (int kf = 0; kf < 32; ++kf) {
      const int k = kf * 4 + hsel * 2;
      const v2f a  = *reinterpret_cast<const v2f*>(&sH[l16 * HSTR + k]);
      const int wr = (kf * 2 + hsel) * WP;
      const v2f w0 = *reinterpret_cast<const v2f*>(&sW[wr + 2 * (n0 + l16)]);
      const v2f w1 = *reinterpret_cast<const v2f*>(&sW[wr + 2 * (n0 + 16 + l16)]);
      acc0 = wmma4(a, w0, acc0);
      acc1 = wmma4(a, w1, acc1);
    }
#pragma unroll
    for (int r = 0; r < 8; ++r) {
      const int m = r + hsel * 8;
      sG[m * GSTR + n0 + l16]      = acc0[r];
      sG[m * GSTR + n0 + 16 + l16] = acc1[r];
    }
    __syncthreads();

    float4 gi = *reinterpret_cast<const float4*>(&sG[ub * GSTR + 0 * H_DIM + uj]);
    float4 gf = *reinterpret_cast<const float4*>(&sG[ub * GSTR + 1 * H_DIM + uj]);
    float4 gg = *reinterpret_cast<const float4*>(&sG[ub * GSTR + 2 * H_DIM + uj]);
    float4 go = *reinterpret_cast<const float4*>(&sG[ub * GSTR + 3 * H_DIM + uj]);
    gi.x += bI.x; gi.y += bI.y; gi.z += bI.z; gi.w += bI.w;
    gf.x += bF.x; gf.y += bF.y; gf.z += bF.z; gf.w += bF.w;
    gg.x += bG.x; gg.y += bG.y; gg.z += bG.z; gg.w += bG.w;
    go.x += bO.x; go.y += bO.y; go.z += bO.z; go.w += bO.w;
    float4 hv;
    {
      float i_ = fast_sigmoid(gi.x), f_ = fast_sigmoid(gf.x);
      float g_ = fast_tanh(gg.x),    o_ = fast_sigmoid(go.x);
      cc0 = f_ * cc0 + i_ * g_;  hv.x = o_ * fast_tanh(cc0);
    }
    {
      float i_ = fast_sigmoid(gi.y), f_ = fast_sigmoid(gf.y);
      float g_ = fast_tanh(gg.y),    o_ = fast_sigmoid(go.y);
      cc1 = f_ * cc1 + i_ * g_;  hv.y = o_ * fast_tanh(cc1);
    }
    {
      float i_ = fast_sigmoid(gi.z), f_ = fast_sigmoid(gf.z);
      float g_ = fast_tanh(gg.z),    o_ = fast_sigmoid(go.z);
      cc2 = f_ * cc2 + i_ * g_;  hv.z = o_ * fast_tanh(cc2);
    }
    {
      float i_ = fast_sigmoid(gi.w), f_ = fast_sigmoid(gf.w);
      float g_ = fast_tanh(gg.w),    o_ = fast_sigmoid(go.w);
      cc3 = f_ * cc3 + i_ * g_;  hv.w = o_ * fast_tanh(cc3);
    }
    *reinterpret_cast<float4*>(&sH[ub * HSTR + uj]) = hv;

    // fused head: out[b,t] = relu(h2) . Wlin + blin  (one wave per batch row)
    float p = fmaxf(hv.x, 0.f) * sWlin[uj + 0] +
              fmaxf(hv.y, 0.f) * sWlin[uj + 1] +
              fmaxf(hv.z, 0.f) * sWlin[uj + 2] +
              fmaxf(hv.w, 0.f) * sWlin[uj + 3];
    p += __shfl_down(p, 16);
    p += __shfl_down(p, 8);
    p += __shfl_down(p, 4);
    p += __shfl_down(p, 2);
    p += __shfl_down(p, 1);
    if (lane == 0) out[(size_t)(b0 + ub) * T_LEN + t] = p + blv;

    // stage h1_{t+1} (async b128)
    if (t + 1 < T_LEN) {
      const int bb = tid >> 5, j = (tid & 31) * 4;
      async_copy_b128(lds_off(&sHin[bb * HSTR + j]),
                      &h1[((size_t)(b0 + bb) * T_LEN + (t + 1)) * H_DIM + j]);
    }
    wait_async0();
    __syncthreads();
  }
}

// ===========================================================================
extern "C" void kernel_launch(void* const* d_in, const int* in_sizes, int n_in,
                              void* d_out, int out_size, void* d_ws,
                              size_t ws_size, hipStream_t stream) {
  (void)in_sizes; (void)n_in; (void)out_size; (void)ws_size;
  const float* x    = (const float*)d_in[0];
  const float* Wih0 = (const float*)d_in[1];
  const float* Whh0 = (const float*)d_in[2];
  const float* bih0 = (const float*)d_in[3];
  const float* bhh0 = (const float*)d_in[4];
  const float* Wih1 = (const float*)d_in[5];
  const float* Whh1 = (const float*)d_in[6];
  const float* bih1 = (const float*)d_in[7];
  const float* bhh1 = (const float*)d_in[8];
  const float* Wlin = (const float*)d_in[9];
  const float* blin = (const float*)d_in[10];
  float* out = (float*)d_out;
  float* h1  = (float*)d_ws;  // B*T*H fp32 = 157.3 MB scratch

  lstm_layer0<<<B_TOT / BT, 512, 0, stream>>>(x, Wih0, Whh0, bih0, bhh0, h1);
  lstm_layer1_out<<<B_TOT / BT, 512, 0, stream>>>(h1, Wih1, Whh1, bih1, bhh1,
                                                  Wlin, blin, out);
}